// TAttn_23871428231839
// MI455X (gfx1250) — compile-verified
//
#include <hip/hip_runtime.h>

typedef float v2f __attribute__((ext_vector_type(2)));
typedef float v8f __attribute__((ext_vector_type(8)));

#define B_ 8
#define T_ 12
#define N_ 128
#define F_ 16
#define D_ 64

__device__ __forceinline__ float fast_exp(float x) {
#if __has_builtin(__builtin_amdgcn_exp2f)
    return __builtin_amdgcn_exp2f(x * 1.4426950408889634f);   // v_exp_f32 is exp2
#else
    return __expf(x);
#endif
}

__device__ __forceinline__ float fast_tanh(float x) {
#if __has_builtin(__builtin_amdgcn_tanhf)
    return __builtin_amdgcn_tanhf(x);                          // gfx1250 v_tanh_f32
#else
    // tanh(x) = 1 - 2/(exp(2x)+1), via exp2 fast path
    float e = fast_exp(2.0f * x);
    return 1.0f - 2.0f / (e + 1.0f);
#endif
}

// One block per (b,t). 256 threads = 8 waves (wave32).
// Phase 1: qp = query[b] @ w_query           (128x64, f32 WMMA)
//          kp = keys[b,t] @ w_key + bias     (128x64, f32 WMMA, bias in C-init)
// Phase 2: scores[i,j] = sum_d tanh(qp[i,d] + kp[j,d]) * trans[d]
//          out[i,j]    = exp(scores[i,j]) / sum_j exp(scores[i,j])   (faithful, unstable)
__global__ __launch_bounds__(256) void TAttn_kernel(
    const float* __restrict__ query,   // [B, N, F]
    const float* __restrict__ keys,    // [B, T, N, F]
    const float* __restrict__ wq,      // [F, D]
    const float* __restrict__ wk,      // [F, D]
    const float* __restrict__ bias,    // [D]
    const float* __restrict__ trans,   // [D]
    float* __restrict__ out)           // [B, T, N, N]
{
    __shared__ float s_qp[N_ * D_];         // stride 64: only broadcast reads
    __shared__ float s_kp[N_ * (D_ + 1)];   // stride 65: conflict-free lane reads
    __shared__ float s_trans[D_];

    const int blk  = blockIdx.x;            // b*T + t
    const int b    = blk / T_;
    const int tid  = threadIdx.x;
    const int lane = tid & 31;
    const int wave = tid >> 5;              // == M-tile index (8 tiles of 16 rows)

    if (tid < D_) s_trans[tid] = trans[tid];

    // ---------------- Phase 1: projection GEMMs with V_WMMA_F32_16X16X4_F32 ----
    const float* Aq = query + (size_t)b   * N_ * F_;   // [128][16]
    const float* Ak = keys  + (size_t)blk * N_ * F_;   // [128][16]

    const int mrow  = lane & 15;
    const int khalf = (lane >> 4) << 1;      // lanes 0-15 -> K{+0,+1}; 16-31 -> K{+2,+3}
    const int m     = wave * 16 + mrow;      // A-fragment row for this lane

    for (int ntile = 0; ntile < 4; ++ntile) {
        const int n = ntile * 16 + mrow;     // B/C-fragment column for this lane
        v8f cq = {0.f, 0.f, 0.f, 0.f, 0.f, 0.f, 0.f, 0.f};
        const float bv = bias[n];            // bias[n] broadcast down the C column
        v8f ck = {bv, bv, bv, bv, bv, bv, bv, bv};
#pragma unroll
        for (int k0 = 0; k0 < F_; k0 += 4) {
            const int k = k0 + khalf;
            v2f aq, ak, bq, bk;
            // A 16x4 layout: lane = M, VGPR v = K (k, k+1) with lane-half giving +2
            aq.x = Aq[m * F_ + k];     aq.y = Aq[m * F_ + k + 1];
            ak.x = Ak[m * F_ + k];     ak.y = Ak[m * F_ + k + 1];
            // B 4x16 layout: lane = N, VGPR v = K row (symmetric split)
            bq.x = wq[k * D_ + n];     bq.y = wq[(k + 1) * D_ + n];
            bk.x = wk[k * D_ + n];     bk.y = wk[(k + 1) * D_ + n];
            cq = __builtin_amdgcn_wmma_f32_16x16x4_f32(false, aq, false, bq,
                                                       (short)0, cq, false, false);
            ck = __builtin_amdgcn_wmma_f32_16x16x4_f32(false, ak, false, bk,
                                                       (short)0, ck, false, false);
        }
        // C/D layout: VGPR r -> M = r (lanes 0-15) / r+8 (lanes 16-31), lane -> N
#pragma unroll
        for (int r = 0; r < 8; ++r) {
            const int row = wave * 16 + r + ((lane >> 4) << 3);
            const int col = ntile * 16 + mrow;
            s_qp[row * D_ + col]       = cq[r];
            s_kp[row * (D_ + 1) + col] = ck[r];
        }
    }
    __syncthreads();

    // ---------------- Phase 2: tanh-dot scores + row-wise exp/sum -------------
    float* oblk = out + (size_t)blk * N_ * N_;
    const int i_end = wave * 16 + 16;
    for (int i = wave * 16; i < i_end; ++i) {
        const float* qrow = &s_qp[i * D_];
        float acc0 = 0.f, acc1 = 0.f, acc2 = 0.f, acc3 = 0.f;
#pragma unroll 8
        for (int d = 0; d < D_; ++d) {
            const float qv = qrow[d];        // wave-uniform broadcast
            const float tv = s_trans[d];     // wave-uniform broadcast
            acc0 += fast_tanh(qv + s_kp[(lane +  0) * (D_ + 1) + d]) * tv;
            acc1 += fast_tanh(qv + s_kp[(lane + 32) * (D_ + 1) + d]) * tv;
            acc2 += fast_tanh(qv + s_kp[(lane + 64) * (D_ + 1) + d]) * tv;
            acc3 += fast_tanh(qv + s_kp[(lane + 96) * (D_ + 1) + d]) * tv;
        }
        float e0 = fast_exp(acc0), e1 = fast_exp(acc1);
        float e2 = fast_exp(acc2), e3 = fast_exp(acc3);
        float s = e0 + e1 + e2 + e3;
#pragma unroll
        for (int off = 16; off > 0; off >>= 1)
            s += __shfl_xor(s, off, 32);     // wave32 butterfly row-sum
        const float inv = 1.0f / s;
        float* orow = oblk + (size_t)i * N_;
        orow[lane +  0] = e0 * inv;
        orow[lane + 32] = e1 * inv;
        orow[lane + 64] = e2 * inv;
        orow[lane + 96] = e3 * inv;
    }
}

extern "C" void kernel_launch(void* const* d_in, const int* in_sizes, int n_in,
                              void* d_out, int out_size, void* d_ws, size_t ws_size,
                              hipStream_t stream) {
    (void)in_sizes; (void)n_in; (void)out_size; (void)d_ws; (void)ws_size;
    const float* query = (const float*)d_in[0];
    const float* keys  = (const float*)d_in[1];
    const float* wq    = (const float*)d_in[2];
    const float* wk    = (const float*)d_in[3];
    const float* bias  = (const float*)d_in[4];
    const float* trans = (const float*)d_in[5];
    float* out = (float*)d_out;

    TAttn_kernel<<<B_ * T_, 256, 0, stream>>>(query, keys, wq, wk, bias, trans, out);
}